// MixConv_52896817217553
// MI455X (gfx1250) — compile-verified
//
#include <hip/hip_runtime.h>
#include <hip/hip_bf16.h>
#include <stdint.h>

// MixConv depthwise conv, fp32, memory-bound (~26us HBM floor on MI455X).
// CDNA5 paths:
//  - interior tiles: TDM tensor_load_to_lds (TENSORcnt) DMA of the 2D haloed
//    tile straight into LDS, issued once per block by wave 0.
//  - border tiles: zero-filled LDS + EXEC-predicated
//    global_load_async_to_lds_b32 (ASYNCcnt) for the in-range region.
// Compute: unrolled KxK fp32 FMA from LDS, non-temporal output stores.

#define THREADS 256
#define TILE_W 32
#define TILE_H 16
#define IMG 112
#define CH_TOTAL 192
#define CH_GROUP 64
#define NBATCH 32

typedef uint32_t v4u __attribute__((ext_vector_type(4)));
typedef uint32_t v8u __attribute__((ext_vector_type(8)));

__device__ __forceinline__ void async_global_to_lds_b32(uint32_t lds_byte_addr,
                                                        const void* gptr) {
  unsigned long long ga = (unsigned long long)gptr;
  asm volatile("global_load_async_to_lds_b32 %0, %1, off"
               :
               : "v"(lds_byte_addr), "v"(ga)
               : "memory");
}

__device__ __forceinline__ void wait_asynccnt0() {
#if __has_builtin(__builtin_amdgcn_s_wait_asynccnt)
  __builtin_amdgcn_s_wait_asynccnt(0);
#else
  asm volatile("s_wait_asynccnt 0x0" ::: "memory");
#endif
}

__device__ __forceinline__ void wait_tensorcnt0() {
#if __has_builtin(__builtin_amdgcn_s_wait_tensorcnt)
  __builtin_amdgcn_s_wait_tensorcnt(0);
#else
  asm volatile("s_wait_tensorcnt 0x0" ::: "memory");
#endif
}

// One 2D tile (tile_h rows x tile_w fp32 elems, row stride in elems) DMA'd
// from global into LDS at lds_base. Caller guarantees the region is fully
// in-bounds; tensor dims are set equal to tile dims so no OOB logic fires.
// D# bitfields per CDNA5 ISA ch.8 (groups 0/1; VADDR2/3 omitted -> NULL).
__device__ __forceinline__ void tdm_load_2d_to_lds(uint32_t lds_base,
                                                   const void* gsrc,
                                                   uint32_t tile_w_elems,
                                                   uint32_t tile_h_rows,
                                                   uint32_t row_stride_elems) {
  const unsigned long long ga = (unsigned long long)gsrc;
  v4u g0;
  g0[0] = 1u;                                 // count=1 (valid descriptor)
  g0[1] = lds_base;                           // lds_addr (bytes)
  g0[2] = (uint32_t)ga;                       // global_addr[31:0]
  g0[3] = ((uint32_t)(ga >> 32) & 0x01FFFFFFu) | (2u << 30); // ga[56:32]|type=2
  v8u g1;
  g1[0] = 2u << 16;                           // workgroup_mask=0, data_size=4B
  g1[1] = tile_w_elems << 16;                 // tensor_dim0[15:0]
  g1[2] = tile_h_rows << 16;                  // tensor_dim0[31:16]=0 | tensor_dim1[15:0]
  g1[3] = tile_w_elems << 16;                 // tensor_dim1[31:16]=0 | tile_dim0
  g1[4] = tile_h_rows;                        // tile_dim1 | tile_dim2=0
  g1[5] = row_stride_elems;                   // tensor_dim0_stride[31:0]
  g1[6] = 0u;                                 // stride0[47:32] | dim1_stride[15:0]
  g1[7] = 0u;                                 // dim1_stride[47:16]
  asm volatile("tensor_load_to_lds %0, %1" : : "s"(g0), "s"(g1) : "memory");
}

template <int K>
__global__ __launch_bounds__(THREADS) void mixconv_dw_kernel(
    const float* __restrict__ x, const float* __restrict__ w,
    float* __restrict__ y, int cbase) {
  constexpr int PAD = (K - 1) / 2;
  constexpr int LW = TILE_W + 2 * PAD;   // LDS tile width  (incl. halo)
  constexpr int LH = TILE_H + 2 * PAD;   // LDS tile height (incl. halo)
  constexpr int LN = LW * LH;

  __shared__ float smem[LN];

  const int tid = threadIdx.x;                 // 0..255
  const int bx  = blockIdx.x;                  // 0..3  (W tiles, last partial)
  const int by  = blockIdx.y;                  // 0..6  (H tiles, exact)
  const int nc  = blockIdx.z;                  // 0..(32*64-1)
  const int cg  = nc & (CH_GROUP - 1);         // channel within group
  const int n   = nc >> 6;                     // batch index
  const int c   = cbase + cg;                  // absolute channel

  const int x0 = bx * TILE_W;
  const int y0 = by * TILE_H;
  const int gx0 = x0 - PAD;
  const int gy0 = y0 - PAD;

  const size_t plane_off = (size_t)(n * CH_TOTAL + c) * (IMG * IMG);
  const float* __restrict__ xplane = x + plane_off;
  float* __restrict__ yplane = y + plane_off;

  const bool interior = (gx0 >= 0) && (gx0 + LW <= IMG) &&
                        (gy0 >= 0) && (gy0 + LH <= IMG);

  if (interior) {
    // ---- TDM path: one 2D descriptor DMA fills the whole LDS tile ----
    // Scalarize the gate so only wave 0 truly branches in (TDM ignores EXEC;
    // an exec-masked skip would issue duplicate DMAs from waves 1..7).
    const int wave_id = __builtin_amdgcn_readfirstlane(tid >> 5);
    if (wave_id == 0) {
      const uint32_t lds_base = (uint32_t)(uintptr_t)(&smem[0]);
      tdm_load_2d_to_lds(lds_base, xplane + gy0 * IMG + gx0,
                         (uint32_t)LW, (uint32_t)LH, (uint32_t)IMG);
      wait_tensorcnt0();
    }
  } else {
    // ---- async path: zero LDS (border zero-padding), then async-copy the
    //      in-range part of the haloed tile ----
    #pragma unroll
    for (int e = tid; e < LN; e += THREADS) smem[e] = 0.0f;
    __syncthreads();  // DS stores done & visible before async writes land

    #pragma unroll 2
    for (int e = tid; e < LN; e += THREADS) {
      const int ly = e / LW;
      const int lx = e - ly * LW;
      const int gy = gy0 + ly;
      const int gx = gx0 + lx;
      if ((unsigned)gy < (unsigned)IMG && (unsigned)gx < (unsigned)IMG) {
        const uint32_t lds_addr = (uint32_t)(uintptr_t)(&smem[e]);
        async_global_to_lds_b32(lds_addr, xplane + gy * IMG + gx);
      }
    }
    wait_asynccnt0();
  }

  // ---- per-channel filter (uniform across the block) ----
  float wk[K * K];
  #pragma unroll
  for (int i = 0; i < K * K; ++i) wk[i] = w[cg * K * K + i];

  __syncthreads();  // LDS tile (TDM or async) visible to all waves

  // ---- compute: 2 output rows per thread, fully unrolled K x K ----
  const int tx  = tid & (TILE_W - 1);  // 0..31
  const int ty0 = tid >> 5;            // 0..7
  const int ox  = x0 + tx;

  if (ox < IMG) {
    #pragma unroll
    for (int r = 0; r < 2; ++r) {
      const int ty = ty0 + r * (TILE_H / 2);
      float acc = 0.0f;
      #pragma unroll
      for (int dy = 0; dy < K; ++dy) {
        #pragma unroll
        for (int dx = 0; dx < K; ++dx) {
          acc = fmaf(smem[(ty + dy) * LW + (tx + dx)], wk[dy * K + dx], acc);
        }
      }
      // output is write-once: stream it past the caches
      __builtin_nontemporal_store(acc, &yplane[(y0 + ty) * IMG + ox]);
    }
  }
}

extern "C" void kernel_launch(void* const* d_in, const int* in_sizes, int n_in,
                              void* d_out, int out_size, void* d_ws,
                              size_t ws_size, hipStream_t stream) {
  (void)in_sizes; (void)n_in; (void)d_ws; (void)ws_size; (void)out_size;

  const float* x  = (const float*)d_in[0];
  const float* w3 = (const float*)d_in[1];
  const float* w5 = (const float*)d_in[2];
  const float* w7 = (const float*)d_in[3];
  float* y = (float*)d_out;

  dim3 grid((IMG + TILE_W - 1) / TILE_W,  // 4
            IMG / TILE_H,                 // 7 (exact)
            NBATCH * CH_GROUP);           // 2048
  dim3 block(THREADS);

  mixconv_dw_kernel<3><<<grid, block, 0, stream>>>(x, w3, y, 0);
  mixconv_dw_kernel<5><<<grid, block, 0, stream>>>(x, w5, y, 64);
  mixconv_dw_kernel<7><<<grid, block, 0, stream>>>(x, w7, y, 128);
}